// HeteroGat_54443005444873
// MI455X (gfx1250) — compile-verified
//
#include <hip/hip_runtime.h>
#include <hip/hip_bf16.h>
#include <math.h>

#define N_NODES 100000
#define NE      800000
#define DIN     128
#define HC      128      // H*C
#define NH      4
#define CC      32
#define BN_EPS  1e-5f

typedef __attribute__((ext_vector_type(2))) float v2f;
typedef __attribute__((ext_vector_type(8))) float v8f;

// ---------------------------------------------------------------------------
// Init: zero d_out accumulator, seg_sum, channel stats; seg_max = -inf
// ---------------------------------------------------------------------------
__global__ __launch_bounds__(256) void init_kernel(float* __restrict__ out,
                                                   float* __restrict__ seg_max,
                                                   float* __restrict__ seg_sum,
                                                   float* __restrict__ chan) {
    const int idx = blockIdx.x * 256 + threadIdx.x;
    if (idx < N_NODES * HC) out[idx] = 0.0f;
    if (idx < N_NODES * NH) {
        seg_max[idx] = -INFINITY;
        seg_sum[idx] = 0.0f;
    }
    if (idx < 256) chan[idx] = 0.0f;
}

// ---------------------------------------------------------------------------
// GEMM: x_l = x@W_l, x_r = x@W_r via V_WMMA_F32_16X16X4_F32.
// Block = 256 threads = 8 waves. Block owns 16 rows; wave w owns cols
// [16w,16w+16). A tile (16x128) staged in LDS, shared by all 8 waves.
// A frag (ISA 7.12.2, 32-bit A 16x4): lanes 0-15 -> K=0,1 ; lanes 16-31 -> K=2,3
// B frag (4x16 row-striped):          VGPR0 -> K=0/2, VGPR1 -> K=1/3
// C/D:   VGPR v -> row v (lanes 0-15) / row v+8 (lanes 16-31)
// ---------------------------------------------------------------------------
__global__ __launch_bounds__(256) void gemm_xlxr(const float* __restrict__ x,
                                                 const float* __restrict__ Wl,
                                                 const float* __restrict__ Wr,
                                                 float* __restrict__ xl,
                                                 float* __restrict__ xr) {
    __shared__ float As[16 * DIN];                       // 8 KB
    const int tid  = threadIdx.x;
    const int row0 = blockIdx.x * 16;

    // cooperative A-tile load: 2048 floats = 512 float4, 2 per thread
    {
        const float4* s = (const float4*)(x + (size_t)row0 * DIN);
        float4*       d = (float4*)As;
        d[tid]       = s[tid];
        d[tid + 256] = s[tid + 256];
    }
    __syncthreads();

    const int wave = tid >> 5;
    const int lane = tid & 31;
    const int col0 = wave * 16;
    const int m    = lane & 15;
    const int kh   = (lane >> 4) * 2;   // 0 for lanes 0-15, 2 for lanes 16-31

    v8f accl = {0.f, 0.f, 0.f, 0.f, 0.f, 0.f, 0.f, 0.f};
    v8f accr = {0.f, 0.f, 0.f, 0.f, 0.f, 0.f, 0.f, 0.f};

    for (int k = 0; k < DIN; k += 4) {
        v2f a, bl, br;
        a.x  = As[m * DIN + k + kh];
        a.y  = As[m * DIN + k + kh + 1];
        bl.x = Wl[(size_t)(k + kh)     * HC + col0 + m];
        bl.y = Wl[(size_t)(k + kh + 1) * HC + col0 + m];
        br.x = Wr[(size_t)(k + kh)     * HC + col0 + m];
        br.y = Wr[(size_t)(k + kh + 1) * HC + col0 + m];
        accl = __builtin_amdgcn_wmma_f32_16x16x4_f32(false, a, false, bl,
                                                     (short)0, accl, false, false);
        accr = __builtin_amdgcn_wmma_f32_16x16x4_f32(false, a, false, br,
                                                     (short)0, accr, false, false);
    }

    const int rbase = row0 + ((lane >> 4) << 3);
#pragma unroll
    for (int v = 0; v < 8; ++v) {
        xl[(size_t)(rbase + v) * HC + col0 + m] = accl[v];
        xr[(size_t)(rbase + v) * HC + col0 + m] = accr[v];
    }
}

// ---------------------------------------------------------------------------
// Float atomic max via signed/unsigned integer ordering trick (init = -inf)
// ---------------------------------------------------------------------------
__device__ __forceinline__ void atomicMaxF(float* addr, float v) {
    if (v >= 0.0f) atomicMax((int*)addr, __float_as_int(v));
    else           atomicMin((unsigned int*)addr, __float_as_uint(v));
}

// ---------------------------------------------------------------------------
// Edge logits: one thread per (edge, head)
// ---------------------------------------------------------------------------
__global__ __launch_bounds__(256) void edge_logits_kernel(
    const float* __restrict__ xl, const float* __restrict__ xr,
    const int* __restrict__ srcI, const int* __restrict__ dstI,
    const float* __restrict__ ew, const float* __restrict__ We,
    const float* __restrict__ att, float* __restrict__ logits,
    float* __restrict__ seg_max) {
    const int idx = blockIdx.x * 256 + threadIdx.x;
    if (idx >= NE * NH) return;
    const int e = idx >> 2;
    const int h = idx & 3;
    const int s = srcI[e];
    const int d = dstI[e];
    const float w = ew[e];

    const float4* pl = (const float4*)(xl + (size_t)s * HC + h * CC);
    const float4* pr = (const float4*)(xr + (size_t)d * HC + h * CC);
    const float4* pe = (const float4*)(We + h * CC);
    const float4* pa = (const float4*)(att + h * CC);

    float acc = 0.0f;
#pragma unroll
    for (int i = 0; i < 8; ++i) {
        float4 a = pl[i], b = pr[i], we = pe[i], at = pa[i];
        float m0 = a.x + b.x + w * we.x; m0 = m0 > 0.f ? m0 : 0.2f * m0;
        float m1 = a.y + b.y + w * we.y; m1 = m1 > 0.f ? m1 : 0.2f * m1;
        float m2 = a.z + b.z + w * we.z; m2 = m2 > 0.f ? m2 : 0.2f * m2;
        float m3 = a.w + b.w + w * we.w; m3 = m3 > 0.f ? m3 : 0.2f * m3;
        acc += at.x * m0 + at.y * m1 + at.z * m2 + at.w * m3;
    }
    logits[idx] = acc;
    atomicMaxF(&seg_max[d * NH + h], acc);
}

// ---------------------------------------------------------------------------
// exp(logit - segmax), accumulate denominator
// ---------------------------------------------------------------------------
__global__ __launch_bounds__(256) void edge_exp_kernel(
    const int* __restrict__ dstI, float* __restrict__ logits,
    const float* __restrict__ seg_max, float* __restrict__ seg_sum) {
    const int idx = blockIdx.x * 256 + threadIdx.x;
    if (idx >= NE * NH) return;
    const int e = idx >> 2;
    const int h = idx & 3;
    const int d = dstI[e];
    const float ex = __expf(logits[idx] - seg_max[d * NH + h]);
    logits[idx] = ex;
    atomicAdd(&seg_sum[d * NH + h], ex);
}

// ---------------------------------------------------------------------------
// scatter: out[dst] += alpha * x_l[src]
// ---------------------------------------------------------------------------
__global__ __launch_bounds__(256) void edge_scatter_kernel(
    const float* __restrict__ xl, const int* __restrict__ srcI,
    const int* __restrict__ dstI, const float* __restrict__ expv,
    const float* __restrict__ seg_sum, float* __restrict__ out) {
    const int idx = blockIdx.x * 256 + threadIdx.x;
    if (idx >= NE * NH) return;
    const int e = idx >> 2;
    const int h = idx & 3;
    const int s = srcI[e];
    const int d = dstI[e];
    const float alpha = expv[idx] / (seg_sum[d * NH + h] + 1e-16f);
    const float* pl = xl + (size_t)s * HC + h * CC;
    float*       po = out + (size_t)d * HC + h * CC;
#pragma unroll
    for (int c = 0; c < CC; ++c) atomicAdd(&po[c], pl[c] * alpha);
}

// ---------------------------------------------------------------------------
// BatchNorm stats: per-channel sum / sumsq (128 threads = 128 channels)
// ---------------------------------------------------------------------------
__global__ __launch_bounds__(128) void bn_stats_kernel(
    const float* __restrict__ out, const float* __restrict__ bias,
    float* __restrict__ chan) {
    const int c = threadIdx.x;
    const float b = bias[c];
    float s = 0.f, s2 = 0.f;
    for (int n = blockIdx.x; n < N_NODES; n += gridDim.x) {
        const float v = out[(size_t)n * HC + c] + b;
        s += v; s2 += v * v;
    }
    atomicAdd(&chan[c], s);
    atomicAdd(&chan[HC + c], s2);
}

__global__ __launch_bounds__(128) void bn_finalize_kernel(
    const float* __restrict__ chan, float* __restrict__ musig) {
    const int c = threadIdx.x;
    const float invN = 1.0f / (float)N_NODES;
    const float mu   = chan[c] * invN;
    const float var  = chan[HC + c] * invN - mu * mu;
    musig[c]      = mu;
    musig[HC + c] = rsqrtf(var + BN_EPS);
}

__global__ __launch_bounds__(256) void bn_apply_kernel(
    float* __restrict__ out, const float* __restrict__ bias,
    const float* __restrict__ gamma, const float* __restrict__ beta,
    const float* __restrict__ musig) {
    const int idx = blockIdx.x * 256 + threadIdx.x;
    if (idx >= N_NODES * HC) return;
    const int c = idx & (HC - 1);
    const float v = out[idx] + bias[c];
    float y = gamma[c] * (v - musig[c]) * musig[HC + c] + beta[c];
    y = y > 0.f ? y : 0.01f * y;
    out[idx] = y;
}

// ---------------------------------------------------------------------------
extern "C" void kernel_launch(void* const* d_in, const int* in_sizes, int n_in,
                              void* d_out, int out_size, void* d_ws, size_t ws_size,
                              hipStream_t stream) {
    const float* x    = (const float*)d_in[0];
    const int*   ei   = (const int*)  d_in[1];   // [2,E] row-major
    const float* ew   = (const float*)d_in[2];
    const float* Wl   = (const float*)d_in[3];
    const float* Wr   = (const float*)d_in[4];
    const float* We   = (const float*)d_in[5];
    const float* att  = (const float*)d_in[6];
    const float* bias = (const float*)d_in[7];
    const float* gam  = (const float*)d_in[8];
    const float* bet  = (const float*)d_in[9];
    float* out = (float*)d_out;

    const int* srcI = ei;
    const int* dstI = ei + NE;

    // workspace layout (floats)
    float* ws = (float*)d_ws;
    float* xl      = ws;                                   // N*128
    float* xr      = xl + (size_t)N_NODES * HC;            // N*128
    float* logits  = xr + (size_t)N_NODES * HC;            // E*4
    float* seg_max = logits + (size_t)NE * NH;             // N*4
    float* seg_sum = seg_max + (size_t)N_NODES * NH;       // N*4
    float* chan    = seg_sum + (size_t)N_NODES * NH;       // 256
    float* musig   = chan + 256;                           // 256

    const int nOut   = N_NODES * HC;                       // 12.8M
    const int nEdgeT = NE * NH;                            // 3.2M

    init_kernel<<<(nOut + 255) / 256, 256, 0, stream>>>(out, seg_max, seg_sum, chan);

    gemm_xlxr<<<N_NODES / 16, 256, 0, stream>>>(x, Wl, Wr, xl, xr);

    edge_logits_kernel<<<(nEdgeT + 255) / 256, 256, 0, stream>>>(
        xl, xr, srcI, dstI, ew, We, att, logits, seg_max);

    edge_exp_kernel<<<(nEdgeT + 255) / 256, 256, 0, stream>>>(
        dstI, logits, seg_max, seg_sum);

    edge_scatter_kernel<<<(nEdgeT + 255) / 256, 256, 0, stream>>>(
        xl, srcI, dstI, logits, seg_sum, out);

    bn_stats_kernel<<<512, 128, 0, stream>>>(out, bias, chan);
    bn_finalize_kernel<<<1, 128, 0, stream>>>(chan, musig);
    bn_apply_kernel<<<(nOut + 255) / 256, 256, 0, stream>>>(out, bias, gam, bet, musig);
}